// EpiGraphModel_6382321401962
// MI455X (gfx1250) — compile-verified
//
#include <hip/hip_runtime.h>
#include <hip/hip_bf16.h>
#include <math.h>

// Problem constants (match reference)
#define B_    2
#define T_    8
#define N_    1000
#define FIN   64
#define HID   128
#define HEADS 2
#define E_    16000
#define G_    (B_ * T_)     // 16 graphs
#define ROWS  (B_ * N_)     // 2000 sequences

typedef float v2f __attribute__((ext_vector_type(2)));
typedef float v8f __attribute__((ext_vector_type(8)));

__device__ __forceinline__ float sigm_(float x) { return 1.0f / (1.0f + __expf(-x)); }

// ---------------------------------------------------------------- fill
__global__ void fill_kernel(float* p, float v, long n) {
    long i = (long)blockIdx.x * blockDim.x + threadIdx.x;
    if (i < n) p[i] = v;
}

// ---------------------------------------------------------------- WMMA f32 GEMM
// C[M,N] = A[M,K] @ B[K,N]   (TB=1: B stored [N,K] row-major, i.e. use B^T)
// One wave computes one 16x16 fp32 tile via V_WMMA_F32_16X16X4_F32, K unrolled
// by 8 (two WMMAs per iteration) with software double-buffered fragments.
//
// Fragment mapping (ISA 7.12.2):
//   A (16x4): lane L -> row L&15, elems {k+2h, k+2h+1}, h = L>>4
//   B (4x16): lane L -> col L&15, elems {k+2h, k+2h+1}
//   D (16x16): VGPR v, lane L -> row v + 8h, col L&15
//
// M-tail handling: A-row m only affects D-row m, so out-of-range rows are
// CLAMPED (loads stay valid, garbage rows are simply not stored). No
// divergence inside the K loop -> EXEC stays all-ones around the WMMAs.
struct frag2 { v2f lo, hi; };   // fragments for k-step and k+4-step

__device__ __forceinline__ frag2 loadA8(const float* __restrict__ row, int k) {
    frag2 f;
    const float* p = row + k;         // row already includes +2h
    f.lo.x = p[0]; f.lo.y = p[1];     // global_load_b64
    f.hi.x = p[4]; f.hi.y = p[5];     // global_load_b64
    return f;
}
__device__ __forceinline__ frag2 loadB_nt(const float* __restrict__ col, int k, int ldb) {
    frag2 f;
    const float* p = col + (long)k * ldb;   // col = Bb + 2h*ldb + ncol
    f.lo.x = p[0];           f.lo.y = p[ldb];
    f.hi.x = p[4 * (long)ldb]; f.hi.y = p[5 * (long)ldb];
    return f;
}

template <int TB>
__global__ void wmma_gemm_f32(const float* __restrict__ A, const float* __restrict__ Bm,
                              float* __restrict__ C,
                              int M, int N, int K, int lda, int ldb, int ldc,
                              long sA, long sB, long sC)
{
    int lane = threadIdx.x;                       // 0..31
    int tn   = blockIdx.x * blockDim.y + threadIdx.y;
    int tm   = blockIdx.y;
    if (tn * 16 >= N || tm * 16 >= M) return;     // wave-uniform

    const float* Ab = A  + (long)blockIdx.z * sA;
    const float* Bb = Bm + (long)blockIdx.z * sB;
    float*       Cb = C  + (long)blockIdx.z * sC;

    int half = lane >> 4;          // 0 or 1
    int l16  = lane & 15;
    int mrow = tm * 16 + l16;
    if (mrow >= M) mrow = M - 1;   // clamp: only affects discarded D rows
    int ncol = tn * 16 + l16;

    const float* Arow = Ab + (long)mrow * lda + 2 * half;
    const float* Bcol = TB ? (Bb + (long)ncol * ldb + 2 * half)
                           : (Bb + (long)(2 * half) * ldb + ncol);

    v8f acc = {0.f, 0.f, 0.f, 0.f, 0.f, 0.f, 0.f, 0.f};

    // K is a multiple of 8 for every GEMM in this model (64/128/256).
    frag2 aC = loadA8(Arow, 0);
    frag2 bC = TB ? loadA8(Bcol, 0) : loadB_nt(Bcol, 0, ldb);
    for (int k = 0; k + 8 < K; k += 8) {
        frag2 aN = loadA8(Arow, k + 8);
        frag2 bN = TB ? loadA8(Bcol, k + 8) : loadB_nt(Bcol, k + 8, ldb);
        acc = __builtin_amdgcn_wmma_f32_16x16x4_f32(false, aC.lo, false, bC.lo,
                                                    (short)0, acc, false, false);
        acc = __builtin_amdgcn_wmma_f32_16x16x4_f32(false, aC.hi, false, bC.hi,
                                                    (short)0, acc, false, false);
        aC = aN; bC = bN;
    }
    acc = __builtin_amdgcn_wmma_f32_16x16x4_f32(false, aC.lo, false, bC.lo,
                                                (short)0, acc, false, false);
    acc = __builtin_amdgcn_wmma_f32_16x16x4_f32(false, aC.hi, false, bC.hi,
                                                (short)0, acc, false, false);

#pragma unroll
    for (int v = 0; v < 8; ++v) {
        int m = tm * 16 + v + half * 8;
        if (m < M) Cb[(long)m * ldc + ncol] = acc[v];
    }
}

// ---------------------------------------------------------------- GAT edge passes
__device__ __forceinline__ void atomicMaxF(float* addr, float val) {
    if (val >= 0.0f) atomicMax((int*)addr, __float_as_int(val));
    else             atomicMin((unsigned int*)addr, __float_as_uint(val));
}

// pass 1: logit[e,k] = sum_j leaky(xl[src]+xr[dst]) * att[k,j]; segment max over dst
__global__ void edge_logit_max(const float* __restrict__ xbase, int xlOff, int xrOff,
                               int rowStride, long gStrideX,
                               const int* __restrict__ src, const int* __restrict__ dst,
                               const float* __restrict__ att, int KH, int H,
                               float* __restrict__ logit, float* __restrict__ segmax)
{
    long tid = (long)blockIdx.x * blockDim.x + threadIdx.x;
    long total = (long)G_ * E_ * KH;
    if (tid >= total) return;
    int  k = (int)(tid % KH);
    long t2 = tid / KH;
    int  e = (int)(t2 % E_);
    int  g = (int)(t2 / E_);
    int  s = src[e], d = dst[e];
    const float* xl = xbase + (long)g * gStrideX + (long)s * rowStride + xlOff + k * H;
    const float* xr = xbase + (long)g * gStrideX + (long)d * rowStride + xrOff + k * H;
    __builtin_prefetch(xl, 0, 3);     // global_prefetch_b8 (rows re-read by many edges)
    __builtin_prefetch(xr, 0, 3);
    const float* a  = att + k * H;
    float acc = 0.f;
    for (int j = 0; j < H; ++j) {
        float v = xl[j] + xr[j];
        v = v > 0.f ? v : 0.2f * v;      // leaky_relu(0.2)
        acc += v * a[j];
    }
    logit[tid] = acc;
    atomicMaxF(&segmax[((long)g * N_ + d) * KH + k], acc);
}

// pass 2: w = exp(logit - m[dst]); den[dst] += w (w overwrites logit)
__global__ void edge_softmax_den(const int* __restrict__ dst, float* __restrict__ logit,
                                 const float* __restrict__ segmax, float* __restrict__ den,
                                 int KH)
{
    long tid = (long)blockIdx.x * blockDim.x + threadIdx.x;
    long total = (long)G_ * E_ * KH;
    if (tid >= total) return;
    int  k = (int)(tid % KH);
    long t2 = tid / KH;
    int  e = (int)(t2 % E_);
    int  g = (int)(t2 / E_);
    int  d = dst[e];
    float w = __expf(logit[tid] - segmax[((long)g * N_ + d) * KH + k]);
    logit[tid] = w;
    atomicAdd(&den[((long)g * N_ + d) * KH + k], w);
}

// pass 3: out[dst, k*H+j] += xl[src, k*H+j] * (w / (den[dst]+1e-16))
__global__ void edge_aggregate(const float* __restrict__ xbase, int xlOff,
                               int rowStride, long gStrideX,
                               const int* __restrict__ src, const int* __restrict__ dst,
                               const float* __restrict__ w, const float* __restrict__ den,
                               float* __restrict__ out, int KH, int H,
                               int outRowStride, long gStrideOut)
{
    long tid = (long)blockIdx.x * blockDim.x + threadIdx.x;
    long total = (long)G_ * E_ * KH * H;
    if (tid >= total) return;
    int  j = (int)(tid % H);
    long t2 = tid / H;
    int  k = (int)(t2 % KH);
    long t3 = t2 / KH;
    int  e = (int)(t3 % E_);
    int  g = (int)(t3 / E_);
    int  s = src[e], d = dst[e];
    float alpha = w[((long)g * E_ + e) * KH + k] /
                  (den[((long)g * N_ + d) * KH + k] + 1e-16f);
    float v = xbase[(long)g * gStrideX + (long)s * rowStride + xlOff + k * H + j] * alpha;
    atomicAdd(&out[(long)g * gStrideOut + (long)d * outRowStride + k * H + j], v);
}

// ---------------------------------------------------------------- bias + BN + ELU (in place)
__global__ void bias_bn_elu(float* __restrict__ h, const float* __restrict__ bias,
                            const float* __restrict__ ga, const float* __restrict__ be,
                            const float* __restrict__ mu, const float* __restrict__ va,
                            int C, long total)
{
    long tid = (long)blockIdx.x * blockDim.x + threadIdx.x;
    if (tid >= total) return;
    int c = (int)(tid % C);
    float v = h[tid] + bias[c];
    v = (v - mu[c]) * rsqrtf(va[c] + 1e-5f) * ga[c] + be[c];
    h[tid] = v > 0.f ? v : (__expf(v) - 1.0f);   // ELU
}

// ---------------------------------------------------------------- GAT2 epilogue + residual -> seq layout
__global__ void spatial_combine(const float* __restrict__ gat2, const float* __restrict__ g2b,
                                const float* __restrict__ bng, const float* __restrict__ bnb,
                                const float* __restrict__ bnm, const float* __restrict__ bnv,
                                const float* __restrict__ xlr1, const float* __restrict__ projb,
                                float* __restrict__ seq)
{
    long tid = (long)blockIdx.x * blockDim.x + threadIdx.x;
    long total = (long)G_ * N_ * HID;
    if (tid >= total) return;
    int  c = (int)(tid % HID);
    long t2 = tid / HID;
    int  n = (int)(t2 % N_);
    int  g = (int)(t2 / N_);
    float v = gat2[tid] + g2b[c];
    v = (v - bnm[c]) * rsqrtf(bnv[c] + 1e-5f) * bng[c] + bnb[c];
    v = v > 0.f ? v : (__expf(v) - 1.0f);
    v += xlr1[(long)g * (N_ * 640L) + (long)n * 640 + 512 + c] + projb[c];
    int b = g / T_, t = g % T_;
    long r = (long)b * N_ + n;
    seq[(r * T_ + t) * HID + c] = v;
}

// ---------------------------------------------------------------- LSTM gate update (one timestep)
__global__ void lstm_gate(const float* __restrict__ xih,  // (ROWS, T, 4*HID)
                          const float* __restrict__ ghh,  // (ROWS, 4*HID)
                          const float* __restrict__ bih, const float* __restrict__ bhh,
                          float* __restrict__ h, float* __restrict__ c,
                          float* __restrict__ hseq, int t)
{
    long tid = (long)blockIdx.x * blockDim.x + threadIdx.x;
    long total = (long)ROWS * HID;
    if (tid >= total) return;
    int j = (int)(tid % HID);
    int r = (int)(tid / HID);
    const float* xr = xih + ((long)r * T_ + t) * (4 * HID);
    const float* gr = ghh + (long)r * (4 * HID);
    float ig = sigm_(xr[j]           + gr[j]           + bih[j]           + bhh[j]);
    float fg = sigm_(xr[j + HID]     + gr[j + HID]     + bih[j + HID]     + bhh[j + HID]);
    float gg = tanhf(xr[j + 2 * HID] + gr[j + 2 * HID] + bih[j + 2 * HID] + bhh[j + 2 * HID]);
    float og = sigm_(xr[j + 3 * HID] + gr[j + 3 * HID] + bih[j + 3 * HID] + bhh[j + 3 * HID]);
    float cv = fg * c[tid] + ig * gg;
    c[tid] = cv;
    float hv = og * tanhf(cv);
    h[tid] = hv;
    if (hseq) hseq[((long)r * T_ + t) * HID + j] = hv;
}

// ---------------------------------------------------------------- FC head: relu(h@W1+b1)@W2+b2
__global__ void fc_head(const float* __restrict__ h, const float* __restrict__ W1,
                        const float* __restrict__ b1, const float* __restrict__ W2,
                        const float* __restrict__ b2, float* __restrict__ out)
{
    int r = blockIdx.x * blockDim.x + threadIdx.x;
    if (r >= ROWS) return;
    const float* hr = h + (long)r * HID;
    float acc = b2[0];
    for (int j = 0; j < HID / 2; ++j) {
        float s = b1[j];
        for (int k = 0; k < HID; ++k) s += hr[k] * W1[k * (HID / 2) + j];
        s = s > 0.f ? s : 0.f;
        acc += s * W2[j];
    }
    out[r] = acc;
}

// ================================================================ host
extern "C" void kernel_launch(void* const* d_in, const int* in_sizes, int n_in,
                              void* d_out, int out_size, void* d_ws, size_t ws_size,
                              hipStream_t stream)
{
    const float* x        = (const float*)d_in[0];
    const int*   ei       = (const int*)  d_in[1];
    const float* proj_W   = (const float*)d_in[2];
    const float* proj_b   = (const float*)d_in[3];
    const float* g1_Wl    = (const float*)d_in[4];
    const float* g1_Wr    = (const float*)d_in[5];
    const float* g1_att   = (const float*)d_in[6];
    const float* g1_b     = (const float*)d_in[7];
    const float* bn1_g    = (const float*)d_in[8];
    const float* bn1_b    = (const float*)d_in[9];
    const float* bn1_m    = (const float*)d_in[10];
    const float* bn1_v    = (const float*)d_in[11];
    const float* g2_Wl    = (const float*)d_in[12];
    const float* g2_Wr    = (const float*)d_in[13];
    const float* g2_att   = (const float*)d_in[14];
    const float* g2_b     = (const float*)d_in[15];
    const float* bn2_g    = (const float*)d_in[16];
    const float* bn2_b    = (const float*)d_in[17];
    const float* bn2_m    = (const float*)d_in[18];
    const float* bn2_v    = (const float*)d_in[19];
    const float* l0_Wih   = (const float*)d_in[20];
    const float* l0_Whh   = (const float*)d_in[21];
    const float* l0_bih   = (const float*)d_in[22];
    const float* l0_bhh   = (const float*)d_in[23];
    const float* l1_Wih   = (const float*)d_in[24];
    const float* l1_Whh   = (const float*)d_in[25];
    const float* l1_bih   = (const float*)d_in[26];
    const float* l1_bhh   = (const float*)d_in[27];
    const float* fc1_W    = (const float*)d_in[28];
    const float* fc1_b    = (const float*)d_in[29];
    const float* fc2_W    = (const float*)d_in[30];
    const float* fc2_b    = (const float*)d_in[31];

    const int* src = ei;
    const int* dst = ei + E_;

    // workspace carve-up (floats)
    float* ws   = (float*)d_ws;
    long   o    = 0;
    float* xlr1 = ws + o; o += (long)G_ * N_ * 640;   // [xl(256) | xr(256) | proj(128)]
    float* gat1 = ws + o; o += (long)G_ * N_ * 256;
    float* xlr2 = ws + o; o += (long)G_ * N_ * 256;   // [xl(128) | xr(128)]
    float* gat2 = ws + o; o += (long)G_ * N_ * 128;
    float* seq  = ws + o; o += (long)ROWS * T_ * HID;
    float* h1s  = ws + o; o += (long)ROWS * T_ * HID;
    float* logit= ws + o; o += (long)G_ * E_ * 2;
    float* segm = ws + o; o += (long)G_ * N_ * 2;
    float* den  = ws + o; o += (long)G_ * N_ * 2;
    float* hbuf = ws + o; o += (long)ROWS * HID;
    float* cbuf = ws + o; o += (long)ROWS * HID;
    float* gates= ws + o; o += (long)ROWS * 4 * HID;
    float* xih  = xlr1;   // aliased: xlr1 is dead once spatial_combine has run

    auto fill = [&](float* p, float v, long n) {
        fill_kernel<<<dim3((unsigned)((n + 255) / 256)), dim3(256), 0, stream>>>(p, v, n);
    };
    auto gemm = [&](const float* A, const float* Bm, float* C, int M, int N, int K,
                    int lda, int ldb, int ldc, long sA, long sB, long sC,
                    int batch, int transB) {
        dim3 blk(32, 4);
        dim3 grd((unsigned)((N / 16 + 3) / 4), (unsigned)((M + 15) / 16), (unsigned)batch);
        if (transB)
            wmma_gemm_f32<1><<<grd, blk, 0, stream>>>(A, Bm, C, M, N, K, lda, ldb, ldc,
                                                      sA, sB, sC);
        else
            wmma_gemm_f32<0><<<grd, blk, 0, stream>>>(A, Bm, C, M, N, K, lda, ldb, ldc,
                                                      sA, sB, sC);
    };

    // ---- node GEMMs for GAT1 + residual projection (batched over 16 graphs)
    gemm(x, g1_Wl,  xlr1 + 0,   N_, 256, FIN, FIN, 256, 640, (long)N_ * FIN, 0, (long)N_ * 640, G_, 0);
    gemm(x, g1_Wr,  xlr1 + 256, N_, 256, FIN, FIN, 256, 640, (long)N_ * FIN, 0, (long)N_ * 640, G_, 0);
    gemm(x, proj_W, xlr1 + 512, N_, 128, FIN, FIN, 128, 640, (long)N_ * FIN, 0, (long)N_ * 640, G_, 0);

    // ---- GAT1 edge softmax + aggregation
    fill(segm, -INFINITY, (long)G_ * N_ * 2);
    fill(den,  0.0f,      (long)G_ * N_ * 2);
    fill(gat1, 0.0f,      (long)G_ * N_ * 256);
    {
        long tot = (long)G_ * E_ * HEADS;
        edge_logit_max<<<dim3((unsigned)((tot + 255) / 256)), dim3(256), 0, stream>>>(
            xlr1, 0, 256, 640, (long)N_ * 640, src, dst, g1_att, HEADS, HID, logit, segm);
        edge_softmax_den<<<dim3((unsigned)((tot + 255) / 256)), dim3(256), 0, stream>>>(
            dst, logit, segm, den, HEADS);
        long tota = tot * HID;
        edge_aggregate<<<dim3((unsigned)((tota + 255) / 256)), dim3(256), 0, stream>>>(
            xlr1, 0, 640, (long)N_ * 640, src, dst, logit, den, gat1, HEADS, HID,
            256, (long)N_ * 256);
    }
    bias_bn_elu<<<dim3((unsigned)(((long)G_ * N_ * 256 + 255) / 256)), dim3(256), 0, stream>>>(
        gat1, g1_b, bn1_g, bn1_b, bn1_m, bn1_v, 256, (long)G_ * N_ * 256);

    // ---- GAT2 node GEMMs
    gemm(gat1, g2_Wl, xlr2 + 0,   N_, 128, 256, 256, 128, 256, (long)N_ * 256, 0, (long)N_ * 256, G_, 0);
    gemm(gat1, g2_Wr, xlr2 + 128, N_, 128, 256, 256, 128, 256, (long)N_ * 256, 0, (long)N_ * 256, G_, 0);

    // ---- GAT2 edge softmax + aggregation (1 head)
    fill(segm, -INFINITY, (long)G_ * N_);
    fill(den,  0.0f,      (long)G_ * N_);
    fill(gat2, 0.0f,      (long)G_ * N_ * 128);
    {
        long tot = (long)G_ * E_;
        edge_logit_max<<<dim3((unsigned)((tot + 255) / 256)), dim3(256), 0, stream>>>(
            xlr2, 0, 128, 256, (long)N_ * 256, src, dst, g2_att, 1, HID, logit, segm);
        edge_softmax_den<<<dim3((unsigned)((tot + 255) / 256)), dim3(256), 0, stream>>>(
            dst, logit, segm, den, 1);
        long tota = tot * HID;
        edge_aggregate<<<dim3((unsigned)((tota + 255) / 256)), dim3(256), 0, stream>>>(
            xlr2, 0, 256, (long)N_ * 256, src, dst, logit, den, gat2, 1, HID,
            128, (long)N_ * 128);
    }

    // ---- BN2 + ELU + residual, reshaped into LSTM sequence layout
    spatial_combine<<<dim3((unsigned)(((long)G_ * N_ * HID + 255) / 256)), dim3(256), 0, stream>>>(
        gat2, g2_b, bn2_g, bn2_b, bn2_m, bn2_v, xlr1, proj_b, seq);

    // ---- LSTM layer 0: hoisted input projection, then 8 sequential steps
    gemm(seq, l0_Wih, xih, ROWS * T_, 4 * HID, HID, HID, HID, 4 * HID, 0, 0, 0, 1, 1);
    fill(hbuf, 0.0f, (long)ROWS * HID);
    fill(cbuf, 0.0f, (long)ROWS * HID);
    for (int t = 0; t < T_; ++t) {
        gemm(hbuf, l0_Whh, gates, ROWS, 4 * HID, HID, HID, HID, 4 * HID, 0, 0, 0, 1, 1);
        lstm_gate<<<dim3((unsigned)(((long)ROWS * HID + 255) / 256)), dim3(256), 0, stream>>>(
            xih, gates, l0_bih, l0_bhh, hbuf, cbuf, h1s, t);
    }

    // ---- LSTM layer 1
    gemm(h1s, l1_Wih, xih, ROWS * T_, 4 * HID, HID, HID, HID, 4 * HID, 0, 0, 0, 1, 1);
    fill(hbuf, 0.0f, (long)ROWS * HID);
    fill(cbuf, 0.0f, (long)ROWS * HID);
    for (int t = 0; t < T_; ++t) {
        gemm(hbuf, l1_Whh, gates, ROWS, 4 * HID, HID, HID, HID, 4 * HID, 0, 0, 0, 1, 1);
        lstm_gate<<<dim3((unsigned)(((long)ROWS * HID + 255) / 256)), dim3(256), 0, stream>>>(
            xih, gates, l1_bih, l1_bhh, hbuf, cbuf, (float*)nullptr, t);
    }

    // ---- FC head -> d_out (B, N, 1) flat = row index b*N+n
    fc_head<<<dim3((ROWS + 255) / 256), dim3(256), 0, stream>>>(
        hbuf, fc1_W, fc1_b, fc2_W, fc2_b, (float*)d_out);
}